// BertTagger1_13469017440314
// MI455X (gfx1250) — compile-verified
//
#include <hip/hip_runtime.h>
#include <cmath>

// ---------------------------------------------------------------------------
// BERT tagger head, fused, fp32 WMMA + async global->LDS staging (gfx1250).
//   seq:   [32, 512, 1024] f32
//   valid: [32, 512] i32
//   W1:    [1024, 512] f32, b1: [512] f32
//   W2:    [512, 4]   f32, b2: [4]  f32
//   out:   [32, 512, 4] f32 = log_softmax(gelu(compact(seq)@W1+b1)@W2+b2)
// ---------------------------------------------------------------------------

typedef __attribute__((ext_vector_type(2))) float v2f;
typedef __attribute__((ext_vector_type(8))) float v8f;

#define BATCH   32
#define SLEN    512
#define HDIM    1024
#define HHALF   512
#define NLAB    4

#define KCHUNK      16
#define NCHUNKS     (HDIM / KCHUNK)   // 64
#define LDSB_STRIDE 520   // 16 x 520 f32 W1 slab (padded vs 512 for banks)
#define H2_STRIDE   516   // 32 x 516 f32 h tile  (padded vs 512 for banks)

#define A_SZ  (32 * KCHUNK)            // 512 floats per A buffer
#define B_SZ  (16 * LDSB_STRIDE)       // 8320 floats per B buffer

// LDS layout (floats):
//  [    0,   512)  A buf 0   (32 x 16)
//  [  512,  1024)  A buf 1
//  [ 1024,  9344)  B buf 0   (16 x 520)
//  [ 9344, 17664)  B buf 1
//  phase 2 reuses [0, 16640): h 32x516 + logits 32x4
//  [17664, 19712)  W2 512x4 (preloaded once, both phases keep clear of it)
#define OFF_A0   0
#define OFF_B0   1024
#define OFF_H    0
#define OFF_L    (32 * H2_STRIDE)
#define OFF_W2   (OFF_B0 + 2 * B_SZ)
#define SMEM_FLOATS (OFF_W2 + HHALF * NLAB)

// ---- CDNA5 async global->LDS DMA (ASYNCcnt path), inline asm -------------
// Flat LDS pointers: addr[31:0] is the wave-relative LDS byte offset
// (see the aperture doc).
__device__ __forceinline__ void async_copy_b128(void* lds_dst, const void* gsrc) {
  unsigned off = (unsigned)(unsigned long long)lds_dst;
  asm volatile("global_load_async_to_lds_b128 %0, %1, off"
               :: "v"(off), "v"(gsrc) : "memory");
}
__device__ __forceinline__ void async_copy_b64(void* lds_dst, const void* gsrc) {
  unsigned off = (unsigned)(unsigned long long)lds_dst;
  asm volatile("global_load_async_to_lds_b64 %0, %1, off"
               :: "v"(off), "v"(gsrc) : "memory");
}
__device__ __forceinline__ void wait_asynccnt0() {
  asm volatile("s_wait_asynccnt 0x0" ::: "memory");
}

// ---------------------------------------------------------------------------
// Kernel 1: stable-compaction gather indices.
// gidx[b][j] = source position of the j-th valid token of row b, else -1.
// ---------------------------------------------------------------------------
__global__ __launch_bounds__(512)
void build_gather_idx(const int* __restrict__ valid, int* __restrict__ gidx) {
  __shared__ int sv[SLEN];
  const int b = blockIdx.x;
  const int t = threadIdx.x;
  const int v = valid[b * SLEN + t];
  sv[t] = v;
  __syncthreads();
#pragma unroll
  for (int off = 1; off < SLEN; off <<= 1) {
    int y = (t >= off) ? sv[t - off] : 0;
    __syncthreads();
    sv[t] += y;
    __syncthreads();
  }
  const int pos = sv[t] - v;  // exclusive prefix sum
  gidx[b * SLEN + t] = -1;    // default: zero-row (tail)
  __syncthreads();
  if (v) gidx[b * SLEN + pos] = t;
}

// ---------------------------------------------------------------------------
// Kernel 2: fused  gather -> GEMM1 (WMMA f32, async double-buffered LDS)
//                 -> GELU -> GEMM2 -> log_softmax
// grid = (16 row-tiles, 32 batches), block = 256 (8 waves of 32).
// ---------------------------------------------------------------------------
__global__ __launch_bounds__(256)
void tagger_fused(const float* __restrict__ seq, const int* __restrict__ gidx,
                  const float* __restrict__ W1, const float* __restrict__ b1,
                  const float* __restrict__ W2, const float* __restrict__ b2,
                  float* __restrict__ out) {
  __shared__ float smem[SMEM_FLOATS];

  const int b    = blockIdx.y;
  const int m0g  = blockIdx.x * 32;        // row block within [0,512)
  const int tid  = threadIdx.x;
  const int lane = tid & 31;
  const int wv   = tid >> 5;               // wave id 0..7
  const int ml   = lane & 15;              // M / N position within tile
  const int kh   = (lane >> 4) * 2;        // K pair select (0 or 2)
  const int nbase = wv * 64;

  // ---- loop-invariant gather state for the A stager ----
  const int arow = tid >> 3;               // 0..31
  const int akk  = (tid & 7) * 2;          // 0,2,..,14
  const int src  = gidx[b * SLEN + m0g + arow];
  const float* aptr = (src >= 0)
      ? seq + ((size_t)b * SLEN + src) * HDIM + akk : (const float*)0;

  // pre-zero invalid rows in both A buffers (async can't zero-fill)
  if (src < 0) {
#pragma unroll
    for (int q = 0; q < 2; q++) {
      float* la = smem + OFF_A0 + q * A_SZ;
      la[arow * KCHUNK + akk]     = 0.0f;
      la[arow * KCHUNK + akk + 1] = 0.0f;
    }
  }
  // preload W2 (512x4) into LDS once: 8 floats per thread
  {
    const int j = tid * 8;
    const float4 w0 = *(const float4*)(W2 + j);
    const float4 w1 = *(const float4*)(W2 + j + 4);
    *(float4*)&smem[OFF_W2 + j]     = w0;
    *(float4*)&smem[OFF_W2 + j + 4] = w1;
  }

  // ---- async stager: chunk ic -> buffer buf (A: 1x b64, B: 8x b128) ----
  auto stage = [&](int ic, int buf) {
    const int kc = ic * KCHUNK;
    float* la = smem + OFF_A0 + buf * A_SZ;
    float* lb = smem + OFF_B0 + buf * B_SZ;
    if (src >= 0)
      async_copy_b64(&la[arow * KCHUNK + akk], aptr + kc);
#pragma unroll
    for (int i = 0; i < 8; i++) {
      const int f  = tid + i * 256;        // float4 id within 16x512 slab
      const int kk = f >> 7;               // 128 float4 per K row
      const int n  = (f & 127) * 4;
      async_copy_b128(&lb[kk * LDSB_STRIDE + n],
                      W1 + (size_t)(kc + kk) * HHALF + n);
    }
  };

  v8f acc[2][4] = {};

  stage(0, 0);
  for (int ic = 0; ic < NCHUNKS; ic++) {
    wait_asynccnt0();      // chunk ic's DMA landed (in-order per wave)
    __syncthreads();       // all waves see it; all done reading other buffer
    if (ic + 1 < NCHUNKS) stage(ic + 1, (ic + 1) & 1);  // overlap next DMA
    const float* la = smem + OFF_A0 + (ic & 1) * A_SZ;
    const float* lb = smem + OFF_B0 + (ic & 1) * B_SZ;

#pragma unroll
    for (int kq = 0; kq < KCHUNK; kq += 4) {
      // A fragments: lane -> M=ml (mt0) / 16+ml (mt1); elems K=kq+kh, +1
      v2f a0, a1;
      a0[0] = la[ml * KCHUNK + kq + kh];
      a0[1] = la[ml * KCHUNK + kq + kh + 1];
      a1[0] = la[(16 + ml) * KCHUNK + kq + kh];
      a1[1] = la[(16 + ml) * KCHUNK + kq + kh + 1];
#pragma unroll
      for (int t = 0; t < 4; t++) {
        const int ncol = nbase + t * 16 + ml;
        v2f bf;  // B rows K=kq+kh (elem0), kq+kh+1 (elem1), col=ncol
        bf[0] = lb[(kq + kh) * LDSB_STRIDE + ncol];
        bf[1] = lb[(kq + kh + 1) * LDSB_STRIDE + ncol];
        acc[0][t] = __builtin_amdgcn_wmma_f32_16x16x4_f32(
            false, a0, false, bf, (short)0, acc[0][t], false, false);
        acc[1][t] = __builtin_amdgcn_wmma_f32_16x16x4_f32(
            false, a1, false, bf, (short)0, acc[1][t], false, false);
      }
    }
  }

  __syncthreads();  // K loop done; safe to overwrite A/B region with h

  // ---- epilogue: +b1, exact GELU, spill h tile to LDS ----
  float* lds_h = smem + OFF_H;
#pragma unroll
  for (int mt = 0; mt < 2; mt++) {
#pragma unroll
    for (int t = 0; t < 4; t++) {
      const int ncol  = nbase + t * 16 + ml;
      const float bias = b1[ncol];
      const int rbase = mt * 16 + 8 * (lane >> 4);
#pragma unroll
      for (int r = 0; r < 8; r++) {
        const float x = acc[mt][t][r] + bias;
        const float g = 0.5f * x * (1.0f + erff(x * 0.70710678118654752f));
        lds_h[(rbase + r) * H2_STRIDE + ncol] = g;
      }
    }
  }
  __syncthreads();

  // ---- GEMM2: logits[row][lab] = h[row,:] . W2[:,lab] + b2[lab] ----
  if (tid < 128) {
    const int row = tid >> 2;
    const int lab = tid & 3;
    float a2 = b2[lab];
    const float* hrow = lds_h + row * H2_STRIDE;
    const float* w2l  = smem + OFF_W2 + lab;
#pragma unroll 8
    for (int n = 0; n < HHALF; n++) a2 += hrow[n] * w2l[n * NLAB];
    smem[OFF_L + row * NLAB + lab] = a2;
  }
  __syncthreads();

  // ---- log_softmax over the 4 labels, write float4 ----
  if (tid < 32) {
    const int row = tid;
    const float x0 = smem[OFF_L + row * NLAB + 0];
    const float x1 = smem[OFF_L + row * NLAB + 1];
    const float x2 = smem[OFF_L + row * NLAB + 2];
    const float x3 = smem[OFF_L + row * NLAB + 3];
    const float m = fmaxf(fmaxf(x0, x1), fmaxf(x2, x3));
    const float lse =
        logf(expf(x0 - m) + expf(x1 - m) + expf(x2 - m) + expf(x3 - m));
    float4 o;
    o.x = x0 - m - lse; o.y = x1 - m - lse;
    o.z = x2 - m - lse; o.w = x3 - m - lse;
    *(float4*)(out + ((size_t)b * SLEN + m0g + row) * NLAB) = o;
  }
}

// ---------------------------------------------------------------------------
extern "C" void kernel_launch(void* const* d_in, const int* in_sizes, int n_in,
                              void* d_out, int out_size, void* d_ws, size_t ws_size,
                              hipStream_t stream) {
  const float* seq   = (const float*)d_in[0];
  const int*   valid = (const int*)d_in[1];
  const float* W1    = (const float*)d_in[2];
  const float* b1    = (const float*)d_in[3];
  const float* W2    = (const float*)d_in[4];
  const float* b2    = (const float*)d_in[5];
  float* out = (float*)d_out;
  int*   gidx = (int*)d_ws;  // [32][512] gather indices (64 KB)

  build_gather_idx<<<BATCH, SLEN, 0, stream>>>(valid, gidx);

  dim3 grid(SLEN / 32, BATCH);  // 16 row-tiles x 32 batches
  tagger_fused<<<grid, 256, 0, stream>>>(seq, gidx, W1, b1, W2, b2, out);
}